// CountingModels_32117765439960
// MI455X (gfx1250) — compile-verified
//
#include <hip/hip_runtime.h>
#include <hip/hip_bf16.h>

typedef __attribute__((ext_vector_type(2))) float v2f;
typedef __attribute__((ext_vector_type(8))) float v8f;

#define BB 4
#define CH 56
#define HH 320
#define WW 320
#define HC 313
#define WC 313
#define NPOSC (HC * WC)        // 97969
#define NCNT (BB * NPOSC)      // 391876
#define NIMG (BB * HH * WW)    // 409600

// ---------------------------------------------------------------------------
// 1) Fused separable 8x8 stride-1 average pool: x[b,c,:,:] -> p[b,c,:,:]
// ---------------------------------------------------------------------------
__global__ void __launch_bounds__(256) avgpool_kernel(const float* __restrict__ x,
                                                      float* __restrict__ p) {
    __shared__ float tin[39][40];
    __shared__ float hs[39][32];
    const int tid = threadIdx.x;
    const int tile = blockIdx.x;          // 0..99
    const int c = blockIdx.y;             // 0..55
    const int b = blockIdx.z;             // 0..3
    const int ty = tile / 10, tx = tile % 10;
    const int bi = ty * 32, bj = tx * 32;
    const float* xp = x + (((size_t)b * CH + c) * HH) * WW;

    for (int t = tid; t < 39 * 39; t += 256) {
        int r = t / 39, cc = t % 39;
        int gi = min(bi + r, HH - 1);
        int gj = min(bj + cc, WW - 1);
        tin[r][cc] = xp[(size_t)gi * WW + gj];
    }
    __syncthreads();
    for (int t = tid; t < 39 * 32; t += 256) {
        int r = t / 32, j = t % 32;
        float s = 0.f;
#pragma unroll
        for (int k = 0; k < 8; ++k) s += tin[r][j + k];
        hs[r][j] = s;
    }
    __syncthreads();
    float* pp = p + (((size_t)b * CH + c) * HC) * WC;
    for (int t = tid; t < 32 * 32; t += 256) {
        int i = t / 32, j = t % 32;
        int oi = bi + i, oj = bj + j;
        if (oi < HC && oj < WC) {
            float s = 0.f;
#pragma unroll
            for (int k = 0; k < 8; ++k) s += hs[i + k][j];
            pp[(size_t)oi * WC + oj] = s * (1.f / 64.f);
        }
    }
}

// ---------------------------------------------------------------------------
// 2) E = exp(mean over channels of x)
// ---------------------------------------------------------------------------
__global__ void chanmean_exp_kernel(const float* __restrict__ x, float* __restrict__ E) {
    int idx = blockIdx.x * blockDim.x + threadIdx.x;
    if (idx >= NIMG) return;
    int hw = idx % (HH * WW);
    int b = idx / (HH * WW);
    const float* xp = x + ((size_t)b * CH) * (HH * WW) + hw;
    float s = 0.f;
#pragma unroll 8
    for (int c = 0; c < CH; ++c) s += xp[(size_t)c * (HH * WW)];
    E[idx] = expf(s * (1.f / 56.f));
}

// ---------------------------------------------------------------------------
// Separable sliding-sum helpers (width/height 8)
// ---------------------------------------------------------------------------
__global__ void hslide_valid_kernel(const float* __restrict__ in, float* __restrict__ out,
                                    int rows, int win) {
    int wout = win - 7;
    int idx = blockIdx.x * blockDim.x + threadIdx.x;
    if (idx >= rows * wout) return;
    int r = idx / wout, j = idx % wout;
    const float* rp = in + (size_t)r * win + j;
    float s = 0.f;
#pragma unroll
    for (int k = 0; k < 8; ++k) s += rp[k];
    out[idx] = s;
}

__global__ void vslide_valid_kernel(const float* __restrict__ in, float* __restrict__ out,
                                    int batches, int hin, int wcols) {
    int hout = hin - 7;
    int idx = blockIdx.x * blockDim.x + threadIdx.x;
    if (idx >= batches * hout * wcols) return;
    int wq = idx % wcols;
    int t = idx / wcols;
    int i = t % hout, b = t / hout;
    const float* cp = in + ((size_t)b * hin + i) * wcols + wq;
    float s = 0.f;
#pragma unroll
    for (int k = 0; k < 8; ++k) s += cp[(size_t)k * wcols];
    out[idx] = s;
}

__global__ void hslide_full_kernel(const float* __restrict__ in, float* __restrict__ out,
                                   int rows, int win) {
    int wout = win + 7;
    int idx = blockIdx.x * blockDim.x + threadIdx.x;
    if (idx >= rows * wout) return;
    int r = idx / wout, q = idx % wout;
    int j0 = max(q - 7, 0), j1 = min(q, win - 1);
    const float* rp = in + (size_t)r * win;
    float s = 0.f;
    for (int j = j0; j <= j1; ++j) s += rp[j];
    out[idx] = s;
}

__global__ void vslide_full_kernel(const float* __restrict__ in, float* __restrict__ out,
                                   int batches, int hin, int wcols) {
    int hout = hin + 7;
    int idx = blockIdx.x * blockDim.x + threadIdx.x;
    if (idx >= batches * hout * wcols) return;
    int wq = idx % wcols;
    int t = idx / wcols;
    int pi = t % hout, b = t / hout;
    int i0 = max(pi - 7, 0), i1 = min(pi, hin - 1);
    const float* cp = in + ((size_t)b * hin) * wcols + wq;
    float s = 0.f;
    for (int i = i0; i <= i1; ++i) s += cp[(size_t)i * wcols];
    out[idx] = s;
}

__global__ void zero_stats_kernel(double* s) {
    if (threadIdx.x < 114) s[threadIdx.x] = 0.0;
}

// ---------------------------------------------------------------------------
// Shared WMMA GEMM core pieces.
// W staged in LDS as float2 (k,k+1) pairs per output channel m so each A
// fragment is a single aligned ds_load_b64 into an even VGPR pair.
// B-fragment loads are branchless: out-of-range columns clamp to n=0 and the
// garbage results are simply never written back (column independence).
// ---------------------------------------------------------------------------
__device__ __forceinline__ void load_w_lds(v2f* Wlds2, const float* __restrict__ w1, int tid) {
    for (int t = tid; t < 28 * 64; t += 256) {
        int kp = t >> 6, m = t & 63;
        v2f val;
        val.x = (m < CH) ? w1[m * CH + 2 * kp] : 0.f;
        val.y = (m < CH) ? w1[m * CH + 2 * kp + 1] : 0.f;
        Wlds2[t] = val;
    }
}

// ---------------------------------------------------------------------------
// 3a) WMMA GEMM pass A: per-channel sum/sumsq of y1 = W1 @ p + b1
// ---------------------------------------------------------------------------
__global__ void __launch_bounds__(256) gemm_stats_kernel(
    const float* __restrict__ p, const float* __restrict__ w1,
    const float* __restrict__ b1, double* __restrict__ gsum, double* __restrict__ gsq) {
    __shared__ v2f Wlds2[28 * 64];
    __shared__ float sb1[64];
    __shared__ float ssum[64], ssq[64];
    const int tid = threadIdx.x;

    load_w_lds(Wlds2, w1, tid);
    if (tid < 64) {
        ssum[tid] = 0.f;
        ssq[tid] = 0.f;
        sb1[tid] = (tid < CH) ? b1[tid] : 0.f;
    }
    __syncthreads();

    const int b = blockIdx.y;
    const int wave = tid >> 5, lane = tid & 31;
    const int half = lane >> 4, l16 = lane & 15;
    const int n = blockIdx.x * 128 + wave * 16 + l16;
    const bool vn = n < NPOSC;
    const float* pcol = p + (size_t)b * CH * NPOSC + (vn ? n : 0);

    v8f acc0 = {}, acc1 = {}, acc2 = {}, acc3 = {};
#pragma unroll
    for (int k0 = 0; k0 < CH; k0 += 4) {
        const int c0 = k0 + (half << 1);
        v2f bf;
        bf.x = pcol[(size_t)c0 * NPOSC];
        bf.y = pcol[(size_t)(c0 + 1) * NPOSC];
        const v2f* wl = &Wlds2[((k0 >> 1) + half) * 64 + l16];
        v2f a0 = wl[0], a1f = wl[16], a2f = wl[32], a3f = wl[48];
        acc0 = __builtin_amdgcn_wmma_f32_16x16x4_f32(false, a0,  false, bf, (short)0, acc0, false, false);
        acc1 = __builtin_amdgcn_wmma_f32_16x16x4_f32(false, a1f, false, bf, (short)0, acc1, false, false);
        acc2 = __builtin_amdgcn_wmma_f32_16x16x4_f32(false, a2f, false, bf, (short)0, acc2, false, false);
        acc3 = __builtin_amdgcn_wmma_f32_16x16x4_f32(false, a3f, false, bf, (short)0, acc3, false, false);
    }

    // butterfly-reduce the 16 columns of each half, then one LDS atomic each
    v8f accs[4] = {acc0, acc1, acc2, acc3};
#pragma unroll
    for (int mt = 0; mt < 4; ++mt)
#pragma unroll
        for (int r = 0; r < 8; ++r) {
            int ch = mt * 16 + r + half * 8;
            if (ch < CH) {
                float v = vn ? (accs[mt][r] + sb1[ch]) : 0.f;
                float v2 = v * v;
#pragma unroll
                for (int m = 1; m < 16; m <<= 1) {
                    v += __shfl_xor(v, m);
                    v2 += __shfl_xor(v2, m);
                }
                if (l16 == 0) {
                    atomicAdd(&ssum[ch], v);
                    atomicAdd(&ssq[ch], v2);
                }
            }
        }
    __syncthreads();
    if (tid < CH) {
        atomicAdd(&gsum[tid], (double)ssum[tid]);
        atomicAdd(&gsq[tid], (double)ssq[tid]);
    }
}

__global__ void finalize_bn1_kernel(const double* __restrict__ gsum, const double* __restrict__ gsq,
                                    const float* __restrict__ g1, const float* __restrict__ be1,
                                    float* __restrict__ a1, float* __restrict__ sh1) {
    int c = threadIdx.x;
    if (c < CH) {
        const double cnt = (double)BB * (double)NPOSC;
        double mean = gsum[c] / cnt;
        double var = gsq[c] / cnt - mean * mean;
        float a = g1[c] * rsqrtf((float)var + 1e-5f);
        a1[c] = a;
        sh1[c] = be1[c] - a * (float)mean;
    }
}

// ---------------------------------------------------------------------------
// 3c) WMMA GEMM pass B: recompute y1, BN1+ReLU, dot with w2 -> z, BN2 stats
// ---------------------------------------------------------------------------
__global__ void __launch_bounds__(256) gemm_apply_kernel(
    const float* __restrict__ p, const float* __restrict__ w1,
    const float* __restrict__ b1, const float* __restrict__ w2,
    const float* __restrict__ b2, const float* __restrict__ a1,
    const float* __restrict__ sh1, float* __restrict__ zbuf,
    double* __restrict__ zsum, double* __restrict__ zsq) {
    __shared__ v2f Wlds2[28 * 64];
    __shared__ float sb1[64], sa1[64], ssh1[64], sw2[64];
    __shared__ float zs[2];
    const int tid = threadIdx.x;

    load_w_lds(Wlds2, w1, tid);
    if (tid < 64) {
        bool v = tid < CH;
        sb1[tid] = v ? b1[tid] : 0.f;
        sa1[tid] = v ? a1[tid] : 0.f;
        ssh1[tid] = v ? sh1[tid] : 0.f;
        sw2[tid] = v ? w2[tid] : 0.f;
    }
    if (tid < 2) zs[tid] = 0.f;
    __syncthreads();

    const int b = blockIdx.y;
    const int wave = tid >> 5, lane = tid & 31;
    const int half = lane >> 4, l16 = lane & 15;
    const int n = blockIdx.x * 128 + wave * 16 + l16;
    const bool vn = n < NPOSC;
    const float* pcol = p + (size_t)b * CH * NPOSC + (vn ? n : 0);

    v8f acc0 = {}, acc1 = {}, acc2 = {}, acc3 = {};
#pragma unroll
    for (int k0 = 0; k0 < CH; k0 += 4) {
        const int c0 = k0 + (half << 1);
        v2f bf;
        bf.x = pcol[(size_t)c0 * NPOSC];
        bf.y = pcol[(size_t)(c0 + 1) * NPOSC];
        const v2f* wl = &Wlds2[((k0 >> 1) + half) * 64 + l16];
        v2f a0 = wl[0], a1f = wl[16], a2f = wl[32], a3f = wl[48];
        acc0 = __builtin_amdgcn_wmma_f32_16x16x4_f32(false, a0,  false, bf, (short)0, acc0, false, false);
        acc1 = __builtin_amdgcn_wmma_f32_16x16x4_f32(false, a1f, false, bf, (short)0, acc1, false, false);
        acc2 = __builtin_amdgcn_wmma_f32_16x16x4_f32(false, a2f, false, bf, (short)0, acc2, false, false);
        acc3 = __builtin_amdgcn_wmma_f32_16x16x4_f32(false, a3f, false, bf, (short)0, acc3, false, false);
    }

    float zpart = 0.f;
    {
        v8f accs[4] = {acc0, acc1, acc2, acc3};
#pragma unroll
        for (int mt = 0; mt < 4; ++mt)
#pragma unroll
            for (int r = 0; r < 8; ++r) {
                int ch = mt * 16 + r + half * 8;
                if (ch < CH) {
                    float v = accs[mt][r] + sb1[ch];
                    float yb = fmaxf(fmaf(sa1[ch], v, ssh1[ch]), 0.f);
                    zpart = fmaf(sw2[ch], yb, zpart);
                }
            }
    }
    // lanes L and L+16 hold the same column n (rows 0-7 vs 8-15): combine
    zpart += __shfl_xor(zpart, 16);
    float myz = (vn && half == 0) ? (zpart + b2[0]) : 0.f;
    if (vn && half == 0) zbuf[(size_t)b * NPOSC + n] = myz;

    // block-level BN2 stats: butterfly over the 16-lane group then LDS atomic
    float zm = myz, zm2 = myz * myz;
#pragma unroll
    for (int m = 1; m < 16; m <<= 1) {
        zm += __shfl_xor(zm, m);
        zm2 += __shfl_xor(zm2, m);
    }
    if (l16 == 0 && half == 0) {
        atomicAdd(&zs[0], zm);
        atomicAdd(&zs[1], zm2);
    }
    __syncthreads();
    if (tid == 0) {
        atomicAdd(zsum, (double)zs[0]);
        atomicAdd(zsq, (double)zs[1]);
    }
}

__global__ void finalize_bn2_kernel(const double* __restrict__ zsum, const double* __restrict__ zsq,
                                    const float* __restrict__ g2, const float* __restrict__ be2,
                                    float* __restrict__ a2sh2) {
    const double cnt = (double)NCNT;
    double mean = zsum[0] / cnt;
    double var = zsq[0] / cnt - mean * mean;
    float a = g2[0] * rsqrtf((float)var + 1e-5f);
    a2sh2[0] = a;
    a2sh2[1] = be2[0] - a * (float)mean;
}

// ---------------------------------------------------------------------------
// 4) C_out = relu(a2*z + sh2) -> d_out; U = C_out / Sexp
// ---------------------------------------------------------------------------
__global__ void count_u_kernel(const float* __restrict__ zbuf, const float* __restrict__ a2sh2,
                               const float* __restrict__ sexp, float* __restrict__ outc,
                               float* __restrict__ U) {
    int idx = blockIdx.x * blockDim.x + threadIdx.x;
    if (idx >= NCNT) return;
    float cv = fmaxf(fmaf(a2sh2[0], zbuf[idx], a2sh2[1]), 0.f);
    outc[idx] = cv;
    U[idx] = cv / sexp[idx];
}

// 5) R = E * V
__global__ void final_r_kernel(const float* __restrict__ E, const float* __restrict__ V,
                               float* __restrict__ outr) {
    int idx = blockIdx.x * blockDim.x + threadIdx.x;
    if (idx >= NIMG) return;
    outr[idx] = E[idx] * V[idx];
}

// ---------------------------------------------------------------------------
extern "C" void kernel_launch(void* const* d_in, const int* in_sizes, int n_in,
                              void* d_out, int out_size, void* d_ws, size_t ws_size,
                              hipStream_t stream) {
    (void)in_sizes; (void)n_in; (void)out_size; (void)ws_size;
    const float* x   = (const float*)d_in[0];
    const float* w1  = (const float*)d_in[1];
    const float* b1  = (const float*)d_in[2];
    const float* w2  = (const float*)d_in[3];
    const float* b2  = (const float*)d_in[4];
    const float* g1  = (const float*)d_in[5];
    const float* be1 = (const float*)d_in[6];
    const float* g2  = (const float*)d_in[7];
    const float* be2 = (const float*)d_in[8];
    float* out = (float*)d_out;

    char* ws = (char*)d_ws;
    size_t off = 0;
    auto alloc = [&](size_t bytes) -> void* {
        void* ptr = ws + off;
        off += (bytes + 255) & ~(size_t)255;
        return ptr;
    };

    float* p     = (float*)alloc((size_t)BB * CH * NPOSC * 4);  // 87.8 MB
    float* E     = (float*)alloc((size_t)NIMG * 4);
    float* He    = (float*)alloc((size_t)BB * HH * WC * 4);
    float* Sexp  = (float*)alloc((size_t)NCNT * 4);
    float* zbuf  = (float*)alloc((size_t)NCNT * 4);
    float* U     = (float*)alloc((size_t)NCNT * 4);
    float* Hu    = (float*)alloc((size_t)BB * HC * WW * 4);
    float* V     = (float*)alloc((size_t)NIMG * 4);
    double* st   = (double*)alloc(114 * 8);
    float* a1    = (float*)alloc(64 * 4);
    float* sh1   = (float*)alloc(64 * 4);
    float* a2sh2 = (float*)alloc(2 * 4);
    double* gsum = st;
    double* gsq  = st + 56;
    double* zsum = st + 112;
    double* zsq  = st + 113;

    const int T = 256;
    avgpool_kernel<<<dim3(100, CH, BB), T, 0, stream>>>(x, p);
    chanmean_exp_kernel<<<(NIMG + T - 1) / T, T, 0, stream>>>(x, E);
    hslide_valid_kernel<<<((BB * HH * WC) + T - 1) / T, T, 0, stream>>>(E, He, BB * HH, WW);
    vslide_valid_kernel<<<(NCNT + T - 1) / T, T, 0, stream>>>(He, Sexp, BB, HH, WC);
    zero_stats_kernel<<<1, 128, 0, stream>>>(st);
    const int gblocks = (NPOSC + 127) / 128;
    gemm_stats_kernel<<<dim3(gblocks, BB), T, 0, stream>>>(p, w1, b1, gsum, gsq);
    finalize_bn1_kernel<<<1, 64, 0, stream>>>(gsum, gsq, g1, be1, a1, sh1);
    gemm_apply_kernel<<<dim3(gblocks, BB), T, 0, stream>>>(p, w1, b1, w2, b2, a1, sh1,
                                                           zbuf, zsum, zsq);
    finalize_bn2_kernel<<<1, 1, 0, stream>>>(zsum, zsq, g2, be2, a2sh2);
    count_u_kernel<<<(NCNT + T - 1) / T, T, 0, stream>>>(zbuf, a2sh2, Sexp, out, U);
    hslide_full_kernel<<<((BB * HC * WW) + T - 1) / T, T, 0, stream>>>(U, Hu, BB * HC, WC);
    vslide_full_kernel<<<(NIMG + T - 1) / T, T, 0, stream>>>(Hu, V, BB, HC, WW);
    final_r_kernel<<<(NIMG + T - 1) / T, T, 0, stream>>>(E, V, out + NCNT);
}